// PatchMatch_66735201845641
// MI455X (gfx1250) — compile-verified
//
#include <hip/hip_runtime.h>
#include <stdint.h>

// PatchMatch offset refinement for B=1, C=32, H=256, W=512, K=9.
// Gather/latency-bound (0.4 GFLOP, 43 MB unique data; rf fits in 192 MB L2).
// No matmul structure (L1 distance) -> WMMA not applicable. CDNA5 paths used:
//  - GLOBAL_LOAD_ASYNC_TO_LDS_B64 (ASYNCcnt) for the layout-transpose pass
//  - s_wait_asynccnt fence, global_prefetch_b8 for offset streams
// Channel-major transposed features make every bilinear tap a coalesced b128.

namespace {
constexpr int Hh = 256;
constexpr int Ww = 512;
constexpr int Cch = 32;
constexpr int Kk = 9;
constexpr int HW = Hh * Ww;
constexpr float TEMP = 10000.0f;
constexpr int TSTRIDE = 66;  // LDS row stride: 8B-aligned rows + conflict-free col reads
}  // namespace

// ---------------------------------------------------------------------------
// Transpose: src [C][H*W] -> dst [H*W][C]. Each block: 32 channels x 64 pixels.
// Phase 1 uses CDNA5 async global->LDS copies (b64 per lane) when available.
// ---------------------------------------------------------------------------
__global__ __launch_bounds__(256) void pm_transpose(const float* __restrict__ src,
                                                    float* __restrict__ dst) {
  __shared__ float tile[Cch][TSTRIDE];
  const int p0 = blockIdx.x * 64;
  const int t = threadIdx.x;

#if defined(__HIP_DEVICE_COMPILE__) && defined(__gfx1250__)
  // Async copy: lane loads 2 floats (b64) from global straight into LDS.
  // GVS addressing: mem = SGPR_base + VGPR_byte_offset; LDS dest addr in VGPR.
#pragma unroll
  for (int i = 0; i < 4; ++i) {
    const int f = t + i * 256;      // float2 index within the 32x64 tile
    const int c = f >> 5;           // channel row (32 float2 per row)
    const int p = (f & 31) * 2;     // pixel pair within row
    const unsigned voff = (unsigned)((c * HW + p0 + p) * 4);      // global byte off
    const unsigned lds = (unsigned)(uintptr_t)(&tile[c][p]);      // LDS byte addr
    asm volatile("global_load_async_to_lds_b64 %0, %1, %2"
                 :
                 : "v"(lds), "v"(voff), "s"(src)
                 : "memory");
  }
  asm volatile("s_wait_asynccnt 0x0" ::: "memory");
#else
#pragma unroll
  for (int i = 0; i < 8; ++i) {
    const int idx = t + i * 256;
    const int c = idx >> 6;
    const int p = idx & 63;
    tile[c][p] = src[c * HW + p0 + p];
  }
#endif
  __syncthreads();

#pragma unroll
  for (int i = 0; i < 8; ++i) {
    const int idx = t + i * 256;
    const int p = idx >> 5;   // pixel
    const int c = idx & 31;   // channel (lane-contiguous -> coalesced store)
    dst[(p0 + p) * Cch + c] = tile[c][p];  // banks (2c+p)%64: conflict-free
  }
}

// ---------------------------------------------------------------------------
// Main pass: one thread per pixel. 9 candidates x 4 bilinear taps x 32 channels.
// USE_RT: right features channel-major (b128 taps). USE_LT: left features too.
// ---------------------------------------------------------------------------
template <bool USE_RT, bool USE_LT>
__global__ __launch_bounds__(256) void pm_eval(const float* __restrict__ lf,
                                               const float* __restrict__ lft,
                                               const float* __restrict__ rf,
                                               const float* __restrict__ rft,
                                               const float* __restrict__ ofx,
                                               const float* __restrict__ ofy,
                                               float* __restrict__ out) {
  const int p = blockIdx.x * 256 + threadIdx.x;  // grid covers HW exactly
  const int y = p >> 9;         // p / 512
  const int x = p & (Ww - 1);   // p % 512

  // Warm L2 for the 9 per-candidate offset streams (gfx1250 global_prefetch_b8).
#pragma unroll
  for (int k = 0; k < Kk; ++k) {
    __builtin_prefetch(ofx + k * HW + p, 0, 1);
    __builtin_prefetch(ofy + k * HW + p, 0, 1);
  }

  // Left features for this pixel: 32 channels in registers.
  float l[Cch];
  if (USE_LT) {
#pragma unroll
    for (int c = 0; c < Cch; c += 4) {
      const float4 v = *(const float4*)(lft + p * Cch + c);
      l[c + 0] = v.x; l[c + 1] = v.y; l[c + 2] = v.z; l[c + 3] = v.w;
    }
  } else {
#pragma unroll
    for (int c = 0; c < Cch; ++c) l[c] = lf[c * HW + p];
  }

  float sc[Kk], oxk[Kk], oyk[Kk];
  const float xf = (float)x, yf = (float)y;

#pragma unroll
  for (int k = 0; k < Kk; ++k) {
    const float ox = ofx[k * HW + p];
    const float oy = ofy[k * HW + p];
    oxk[k] = ox;
    oyk[k] = oy;

    // Replicate reference arithmetic: clip -> grid-normalize -> unnormalize.
    const float rx = fminf(fmaxf(xf - ox, 0.0f), (float)(Ww - 1));
    const float ry = fminf(fmaxf(yf - oy, 0.0f), (float)(Hh - 1));
    const float gx = (rx - (float)(Ww / 2)) * (1.0f / (float)(Ww / 2));
    const float gy = (ry - (float)(Hh / 2)) * (1.0f / (float)(Hh / 2));
    const float ix = ((gx + 1.0f) * (float)Ww - 1.0f) * 0.5f;
    const float iy = ((gy + 1.0f) * (float)Hh - 1.0f) * 0.5f;

    const float x0f = floorf(ix), y0f = floorf(iy);
    const float wx = ix - x0f, wy = iy - y0f;
    const int x0 = (int)x0f, y0 = (int)y0f;
    const int x1 = x0 + 1, y1 = y0 + 1;

    const bool vx0 = (x0 >= 0) && (x0 < Ww);
    const bool vx1 = (x1 >= 0) && (x1 < Ww);
    const bool vy0 = (y0 >= 0) && (y0 < Hh);
    const bool vy1 = (y1 >= 0) && (y1 < Hh);
    const int xi0 = min(max(x0, 0), Ww - 1), xi1 = min(max(x1, 0), Ww - 1);
    const int yi0 = min(max(y0, 0), Hh - 1), yi1 = min(max(y1, 0), Hh - 1);

    // Zero-padding folded into the bilinear weights.
    const float w00 = (vx0 && vy0) ? (1.0f - wx) * (1.0f - wy) : 0.0f;
    const float w01 = (vx1 && vy0) ? wx * (1.0f - wy) : 0.0f;
    const float w10 = (vx0 && vy1) ? (1.0f - wx) * wy : 0.0f;
    const float w11 = (vx1 && vy1) ? wx * wy : 0.0f;

    float acc = 0.0f;
    if (USE_RT) {
      const int b00 = (yi0 * Ww + xi0) * Cch;
      const int b01 = (yi0 * Ww + xi1) * Cch;
      const int b10 = (yi1 * Ww + xi0) * Cch;
      const int b11 = (yi1 * Ww + xi1) * Cch;
#pragma unroll
      for (int c = 0; c < Cch; c += 4) {
        const float4 a = *(const float4*)(rft + b00 + c);
        const float4 b = *(const float4*)(rft + b01 + c);
        const float4 u = *(const float4*)(rft + b10 + c);
        const float4 d = *(const float4*)(rft + b11 + c);
        const float s0 = a.x * w00 + b.x * w01 + u.x * w10 + d.x * w11;
        const float s1 = a.y * w00 + b.y * w01 + u.y * w10 + d.y * w11;
        const float s2 = a.z * w00 + b.z * w01 + u.z * w10 + d.z * w11;
        const float s3 = a.w * w00 + b.w * w01 + u.w * w10 + d.w * w11;
        acc += fabsf(l[c + 0] - s0);
        acc += fabsf(l[c + 1] - s1);
        acc += fabsf(l[c + 2] - s2);
        acc += fabsf(l[c + 3] - s3);
      }
    } else {
      const int i00 = yi0 * Ww + xi0, i01 = yi0 * Ww + xi1;
      const int i10 = yi1 * Ww + xi0, i11 = yi1 * Ww + xi1;
#pragma unroll
      for (int c = 0; c < Cch; ++c) {
        const int cb = c * HW;
        const float s0 = rf[cb + i00] * w00 + rf[cb + i01] * w01 +
                         rf[cb + i10] * w10 + rf[cb + i11] * w11;
        acc += fabsf(l[c] - s0);
      }
    }
    sc[k] = -(acc * (1.0f / (float)Cch)) * TEMP;  // mean(-|diff|) * T
  }

  // Softmax over K=9 (max-subtracted, like jax.nn.softmax), weighted offset mean.
  float m = sc[0];
#pragma unroll
  for (int k = 1; k < Kk; ++k) m = fmaxf(m, sc[k]);
  float sw = 0.0f, ax = 0.0f, ay = 0.0f;
#pragma unroll
  for (int k = 0; k < Kk; ++k) {
    const float w = __expf(sc[k] - m);
    sw += w;
    ax += oxk[k] * w;
    ay += oyk[k] * w;
  }
  const float inv = 1.0f / sw;
  out[p] = ax * inv;        // ox_out [1,1,H,W]
  out[HW + p] = ay * inv;   // oy_out [1,1,H,W]
}

// ---------------------------------------------------------------------------
extern "C" void kernel_launch(void* const* d_in, const int* in_sizes, int n_in,
                              void* d_out, int out_size, void* d_ws, size_t ws_size,
                              hipStream_t stream) {
  (void)in_sizes; (void)n_in; (void)out_size;
  const float* lf = (const float*)d_in[0];   // [1,32,256,512]
  const float* rf = (const float*)d_in[1];   // [1,32,256,512]
  const float* ox = (const float*)d_in[2];   // [1,9,256,512]
  const float* oy = (const float*)d_in[3];   // [1,9,256,512]
  float* out = (float*)d_out;                // [2*256*512]

  const size_t need = (size_t)HW * Cch * sizeof(float);  // 16.8 MB each
  if (ws_size >= 2 * need) {
    float* rft = (float*)d_ws;
    float* lft = (float*)((char*)d_ws + need);
    pm_transpose<<<HW / 64, 256, 0, stream>>>(rf, rft);
    pm_transpose<<<HW / 64, 256, 0, stream>>>(lf, lft);
    pm_eval<true, true><<<HW / 256, 256, 0, stream>>>(lf, lft, rf, rft, ox, oy, out);
  } else if (ws_size >= need) {
    float* rft = (float*)d_ws;
    pm_transpose<<<HW / 64, 256, 0, stream>>>(rf, rft);
    pm_eval<true, false><<<HW / 256, 256, 0, stream>>>(lf, nullptr, rf, rft, ox, oy, out);
  } else {
    pm_eval<false, false><<<HW / 256, 256, 0, stream>>>(lf, nullptr, rf, nullptr, ox, oy, out);
  }
}